// SelfAttentionBlock_37271726195167
// MI455X (gfx1250) — compile-verified
//
#include <hip/hip_runtime.h>

// Problem constants (match reference)
#define BQ     4
#define CIN    64
#define HH     96
#define WW     96
#define GROUPS 8
#define NCH    8     // out channels per group
#define CGC    8     // in channels per group
#define KW     7
#define PADR   3
#define HWSZ   (HH * WW)

#define ROWS   4               // output rows per block
#define KVROWS (ROWS + 2*PADR) // k/v halo rows per block = 10
#define NTHR   128             // 4 waves

typedef __attribute__((ext_vector_type(2))) float v2f;
typedef __attribute__((ext_vector_type(4))) float v4f;
typedef __attribute__((ext_vector_type(8))) float v8f;

__device__ __forceinline__ int reflect96(int i) {
    if (i < 0)  i = -i;
    if (i > 95) i = 190 - i;
    return i;
}

// One block = (batch b, group g, 4 output rows). 128 threads = 4 waves.
// Phase A: WMMA f32 16x16x4 grouped 1x1 conv -> k,v for 10 halo rows + q for 4 rows,
//          stored channel-contiguous in LDS ([row][col][8ch]) via b128 stores.
// Phase B: per-pixel online-softmax attention over the 7x7 window, b128 LDS reads,
//          position bias hoisted out of the inner dot (row bias for g<4, col bias for g>=4).
__global__ __launch_bounds__(NTHR)
void SelfAttentionBlock_kernel(const float* __restrict__ x,
                               const float* __restrict__ Wk,
                               const float* __restrict__ Wq,
                               const float* __restrict__ Wv,
                               const float* __restrict__ r_ai,
                               const float* __restrict__ r_aj,
                               float* __restrict__ out)
{
    __shared__ __attribute__((aligned(16))) float kb[KVROWS][WW][NCH];
    __shared__ __attribute__((aligned(16))) float vb[KVROWS][WW][NCH];
    __shared__ __attribute__((aligned(16))) float qb[ROWS][WW][NCH];
    __shared__ __attribute__((aligned(16))) float rcomp[KW][NCH]; // bias vectors (ki for g<4, kj for g>=4)

    const int h0   = blockIdx.x * ROWS;
    const int g    = blockIdx.y;
    const int b    = blockIdx.z;
    const int tid  = threadIdx.x;
    const int lane = tid & 31;
    const int wave = tid >> 5;        // 0..3
    const int half = lane >> 4;       // 0: lanes 0-15, 1: lanes 16-31
    const int m    = lane & 15;       // A-matrix row / B-matrix pixel column

    // ---- stage relative-position bias vectors ----
    // g<4: bias depends only on ki -> rcomp[ki][n] = r_ai[(g*8+n)*7 + ki]
    // g>=4: bias depends only on kj -> rcomp[kj][n] = r_aj[(g*8+n-32)*7 + kj]
    if (tid < KW * NCH) {
        int t = tid / NCH, n = tid % NCH;
        int cg = g * NCH + n;
        rcomp[t][n] = (cg < 32) ? r_ai[cg * KW + t] : r_aj[(cg - 32) * KW + t];
    }

    // ---- build A matrices (weights) in WMMA 16x16x4 layout ----
    // A VGPR0 holds K = (half?2:0), VGPR1 holds K = (half?3:1); M = lane&15.
    const float* WkG = Wk + g * 64;   // [n][i] 8x8
    const float* WqG = Wq + g * 64;
    const float* WvG = Wv + g * 64;
    const int k_lo = half * 2;        // K chunk 0..3
    const int k_hi = 4 + half * 2;    // K chunk 4..7

    v2f aKV_lo, aKV_hi, aQ_lo, aQ_hi;
    if (m < 8) {  // D rows 0-7 = k channels
        aKV_lo[0] = WkG[m * 8 + k_lo];     aKV_lo[1] = WkG[m * 8 + k_lo + 1];
        aKV_hi[0] = WkG[m * 8 + k_hi];     aKV_hi[1] = WkG[m * 8 + k_hi + 1];
    } else {      // D rows 8-15 = v channels
        aKV_lo[0] = WvG[(m - 8) * 8 + k_lo]; aKV_lo[1] = WvG[(m - 8) * 8 + k_lo + 1];
        aKV_hi[0] = WvG[(m - 8) * 8 + k_hi]; aKV_hi[1] = WvG[(m - 8) * 8 + k_hi + 1];
    }
    // q: duplicate Wq in rows 8-15 (upper-half result is a copy, ignored)
    aQ_lo[0] = WqG[(m & 7) * 8 + k_lo];  aQ_lo[1] = WqG[(m & 7) * 8 + k_lo + 1];
    aQ_hi[0] = WqG[(m & 7) * 8 + k_hi];  aQ_hi[1] = WqG[(m & 7) * 8 + k_hi + 1];

    const float* xg = x + (size_t)(b * CIN + g * CGC) * HWSZ;

    // ---- Phase A1: k & v for KVROWS halo rows, 6 strips of 16 px = 60 tiles ----
    for (int t = wave; t < KVROWS * 6; t += 4) {
        int i     = t / 6;
        int strip = (t % 6) * 16;
        int hr    = reflect96(h0 - PADR + i);
        int col   = strip + m;
        const float* xp = xg + hr * WW + col;
        v2f blo, bhi;  // B element (K=input channel, N=pixel)
        blo[0] = xp[(size_t)k_lo       * HWSZ];
        blo[1] = xp[(size_t)(k_lo + 1) * HWSZ];
        bhi[0] = xp[(size_t)k_hi       * HWSZ];
        bhi[1] = xp[(size_t)(k_hi + 1) * HWSZ];

        v8f acc = {};
        acc = __builtin_amdgcn_wmma_f32_16x16x4_f32(false, aKV_lo, false, blo,
                                                    (short)0, acc, false, false);
        acc = __builtin_amdgcn_wmma_f32_16x16x4_f32(false, aKV_hi, false, bhi,
                                                    (short)0, acc, false, false);
        // D VGPR r: lanes 0-15 -> k channel r, lanes 16-31 -> v channel r (M=r+8)
        float* dst = half ? &vb[i][col][0] : &kb[i][col][0];
        v4f lo4 = {acc[0], acc[1], acc[2], acc[3]};
        v4f hi4 = {acc[4], acc[5], acc[6], acc[7]};
        *(v4f*)dst       = lo4;
        *((v4f*)dst + 1) = hi4;
    }

    // ---- Phase A2: q for ROWS center rows, 24 tiles ----
    for (int t = wave; t < ROWS * 6; t += 4) {
        int rr    = t / 6;
        int strip = (t % 6) * 16;
        int col   = strip + m;
        const float* xp = xg + (h0 + rr) * WW + col;
        v2f blo, bhi;
        blo[0] = xp[(size_t)k_lo       * HWSZ];
        blo[1] = xp[(size_t)(k_lo + 1) * HWSZ];
        bhi[0] = xp[(size_t)k_hi       * HWSZ];
        bhi[1] = xp[(size_t)(k_hi + 1) * HWSZ];

        v8f acc = {};
        acc = __builtin_amdgcn_wmma_f32_16x16x4_f32(false, aQ_lo, false, blo,
                                                    (short)0, acc, false, false);
        acc = __builtin_amdgcn_wmma_f32_16x16x4_f32(false, aQ_hi, false, bhi,
                                                    (short)0, acc, false, false);
        if (!half) {
            float* dst = &qb[rr][col][0];
            v4f lo4 = {acc[0], acc[1], acc[2], acc[3]};
            v4f hi4 = {acc[4], acc[5], acc[6], acc[7]};
            *(v4f*)dst       = lo4;
            *((v4f*)dst + 1) = hi4;
        }
    }

    __syncthreads();

    // ---- Phase B: attention; 384 pixels over 128 threads (3 per thread) ----
    const bool rowBias = (g < 4);
    for (int p = tid; p < ROWS * WW; p += NTHR) {
        const int rr = p / WW;     // row within tile (uniform per wave-iteration)
        const int w  = p % WW;

        const v4f q0 = *(const v4f*)&qb[rr][w][0];
        const v4f q1 = *(const v4f*)&qb[rr][w][4];

        // hoisted bias dots: rb[t] = sum_n q[n] * r[t][n]
        float rb[KW];
        #pragma unroll
        for (int t = 0; t < KW; ++t) {
            const v4f r0 = *(const v4f*)&rcomp[t][0];
            const v4f r1 = *(const v4f*)&rcomp[t][4];
            float s = 0.f;
            #pragma unroll
            for (int n = 0; n < 4; ++n) s = __builtin_fmaf(q0[n], r0[n], s);
            #pragma unroll
            for (int n = 0; n < 4; ++n) s = __builtin_fmaf(q1[n], r1[n], s);
            rb[t] = s;
        }

        float mmax = -3.0e38f, ssum = 0.0f;
        float acc[NCH];
        #pragma unroll
        for (int n = 0; n < NCH; ++n) acc[n] = 0.0f;

        for (int ki = 0; ki < KW; ++ki) {
            const int slot = rr + ki;   // kb slot rr+ki == feature row (h0+rr)-3+ki
            for (int kj = 0; kj < KW; ++kj) {
                int c = w + kj - PADR;
                c = (c < 0) ? -c : ((c > 95) ? 190 - c : c);

                const v4f k0 = *(const v4f*)&kb[slot][c][0];
                const v4f k1 = *(const v4f*)&kb[slot][c][4];
                float l = rb[rowBias ? ki : kj];
                #pragma unroll
                for (int n = 0; n < 4; ++n) l = __builtin_fmaf(q0[n], k0[n], l);
                #pragma unroll
                for (int n = 0; n < 4; ++n) l = __builtin_fmaf(q1[n], k1[n], l);

                const float mn   = fmaxf(mmax, l);
                const float corr = __expf(mmax - mn);
                const float e    = __expf(l - mn);
                ssum = ssum * corr + e;

                const v4f va = *(const v4f*)&vb[slot][c][0];
                const v4f vbb = *(const v4f*)&vb[slot][c][4];
                #pragma unroll
                for (int n = 0; n < 4; ++n) acc[n]     = acc[n]     * corr + e * va[n];
                #pragma unroll
                for (int n = 0; n < 4; ++n) acc[n + 4] = acc[n + 4] * corr + e * vbb[n];
                mmax = mn;
            }
        }

        const float inv = 1.0f / ssum;
        float* op = out + ((size_t)(b * CIN + g * NCH) * HH + (h0 + rr)) * WW + w;
        #pragma unroll
        for (int n = 0; n < NCH; ++n)
            op[(size_t)n * HWSZ] = acc[n] * inv;
    }
}

extern "C" void kernel_launch(void* const* d_in, const int* in_sizes, int n_in,
                              void* d_out, int out_size, void* d_ws, size_t ws_size,
                              hipStream_t stream) {
    (void)in_sizes; (void)n_in; (void)out_size; (void)d_ws; (void)ws_size;
    const float* x    = (const float*)d_in[0];
    const float* Wk   = (const float*)d_in[1];
    const float* Wq   = (const float*)d_in[2];
    const float* Wv   = (const float*)d_in[3];
    const float* r_ai = (const float*)d_in[4];
    const float* r_aj = (const float*)d_in[5];
    float* out = (float*)d_out;

    dim3 grid(HH / ROWS, GROUPS, BQ);   // (row tile, group, batch)
    SelfAttentionBlock_kernel<<<grid, NTHR, 0, stream>>>(x, Wk, Wq, Wv, r_ai, r_aj, out);
}